// NCNPredictor_77292231459355
// MI455X (gfx1250) — compile-verified
//
#include <hip/hip_runtime.h>
#include <stdint.h>

#define DIN  128
#define DHID 512
#define CAP  512   // per-target adjacency list capacity (mean degree = 64, Poisson)

typedef __attribute__((ext_vector_type(2))) float v2f;
typedef __attribute__((ext_vector_type(8))) float v8f;

// ---------------------------------------------------------------- init maps/counts
__global__ void ncn_init(int* __restrict__ map_i, int* __restrict__ map_j,
                         int* __restrict__ cnt_i, int* __restrict__ cnt_j,
                         int n, int b) {
    int i = blockIdx.x * blockDim.x + threadIdx.x;
    if (i < n) { map_i[i] = -1; map_j[i] = -1; }
    if (i < b) { cnt_i[i] = 0;  cnt_j[i] = 0;  }
}

__global__ void ncn_setmap(const long long* __restrict__ tar,
                           int* __restrict__ map_i, int* __restrict__ map_j, int B) {
    int b = blockIdx.x * blockDim.x + threadIdx.x;
    if (b < B) {
        map_i[(int)tar[b]]     = b;   // targets are unique (permutation slice)
        map_j[(int)tar[B + b]] = b;
    }
}

// ------------------------------------------- scatter symmetrized edges into lists
__global__ void ncn_scatter(const long long* __restrict__ ei, int E,
                            const int* __restrict__ map_i, const int* __restrict__ map_j,
                            int* __restrict__ cnt_i, int* __restrict__ cnt_j,
                            int* __restrict__ list_i, int* __restrict__ list_j) {
    int e = blockIdx.x * blockDim.x + threadIdx.x;
    if (e >= E) return;
    int u = (int)ei[e];
    int v = (int)ei[E + e];
    int b;
    // direction u->v (neighbor v of u) and v->u (neighbor u of v), both map sides
    b = map_i[u]; if (b >= 0) { int p = atomicAdd(&cnt_i[b], 1); if (p < CAP) list_i[b * CAP + p] = v; }
    b = map_i[v]; if (b >= 0) { int p = atomicAdd(&cnt_i[b], 1); if (p < CAP) list_i[b * CAP + p] = u; }
    b = map_j[u]; if (b >= 0) { int p = atomicAdd(&cnt_j[b], 1); if (p < CAP) list_j[b * CAP + p] = v; }
    b = map_j[v]; if (b >= 0) { int p = atomicAdd(&cnt_j[b], 1); if (p < CAP) list_j[b * CAP + p] = u; }
}

// ---------------------- per-pair: intersect neighbor lists, build xs = [xij | xcn]
__global__ __launch_bounds__(128)
void ncn_pairs(const float* __restrict__ x, const long long* __restrict__ tar,
               const int* __restrict__ cnt_i, const int* __restrict__ cnt_j,
               const int* __restrict__ list_i, const int* __restrict__ list_j,
               float* __restrict__ xs, int B) {
    __shared__ int sLi[CAP];
    __shared__ int sLj[CAP];
    __shared__ int sW[CAP];

    int b   = blockIdx.x;
    int t   = threadIdx.x;          // 0..127 == feature dim
    int n_i = min(cnt_i[b], CAP);
    int n_j = min(cnt_j[b], CAP);

    for (int a = t; a < n_i; a += 128) sLi[a] = list_i[b * CAP + a];
    for (int a = t; a < n_j; a += 128) sLj[a] = list_j[b * CAP + a];
    __syncthreads();

    // weight of entry a in list_i = multiplicity of that node in list_j
    for (int a = t; a < n_i; a += 128) {
        int v = sLi[a], w = 0;
        for (int c = 0; c < n_j; ++c) w += (v == sLj[c]) ? 1 : 0;
        sW[a] = w;
    }
    __syncthreads();

    // xcn[b, t] = sum_a w[a] * x[Li[a], t]   (duplicates in Li carry mult_i)
    float acc = 0.0f;
    for (int a = 0; a < n_i; ++a) {
        int w = sW[a];
        if (w > 0) acc += (float)w * x[(size_t)sLi[a] * DIN + t];
    }

    int inode = (int)tar[b];
    int jnode = (int)tar[B + b];
    float xij = x[(size_t)inode * DIN + t] * x[(size_t)jnode * DIN + t];

    xs[(size_t)b * (2 * DIN) + t]       = xij;
    xs[(size_t)b * (2 * DIN) + DIN + t] = acc;
}

// ---------------- GEMM1: h = relu(xs[1024,256] @ W1[256,512] + b1), f32 WMMA 16x16x4
__global__ __launch_bounds__(32)
void ncn_gemm1_wmma(const float* __restrict__ xs, const float* __restrict__ W1,
                    const float* __restrict__ b1, float* __restrict__ h) {
    const int m0   = blockIdx.x * 16;        // 1024/16 = 64
    const int n0   = blockIdx.y * 16;        // 512/16  = 32
    const int lane = threadIdx.x;            // one wave32 per block
    const int half = lane >> 4;              // 0: K pair {0,1} / M 0..7 ; 1: K {2,3} / M 8..15
    const int l    = lane & 15;

    // A 16x4 f32: lanes 0-15 hold row m0+l K={k,k+1}; lanes 16-31 hold K={k+2,k+3}
    const float* arow = xs + (size_t)(m0 + l) * (2 * DIN) + 2 * half;

    v8f c0 = {};
    v8f c1 = {};
    #pragma unroll 4
    for (int k = 0; k < 2 * DIN; k += 8) {
        // --- step k..k+3 into c0 ---
        v2f a0 = *(const v2f*)(arow + k);
        v2f b0;
        b0.x = W1[(size_t)(k + 2 * half + 0) * DHID + n0 + l];
        b0.y = W1[(size_t)(k + 2 * half + 1) * DHID + n0 + l];
        c0 = __builtin_amdgcn_wmma_f32_16x16x4_f32(false, a0, false, b0,
                                                   (short)0, c0, false, false);
        // --- step k+4..k+7 into c1 (independent accumulator hides WMMA latency) ---
        v2f a1 = *(const v2f*)(arow + k + 4);
        v2f b1v;
        b1v.x = W1[(size_t)(k + 4 + 2 * half + 0) * DHID + n0 + l];
        b1v.y = W1[(size_t)(k + 4 + 2 * half + 1) * DHID + n0 + l];
        c1 = __builtin_amdgcn_wmma_f32_16x16x4_f32(false, a1, false, b1v,
                                                   (short)0, c1, false, false);
    }

    const int col  = n0 + l;
    const float bv = b1[col];
    #pragma unroll
    for (int r = 0; r < 8; ++r) {
        int row = m0 + r + 8 * half;         // C/D: VGPR r -> M=r (lanes 0-15), M=r+8 (16-31)
        float v = c0[r] + c1[r] + bv;
        h[(size_t)row * DHID + col] = v > 0.0f ? v : 0.0f;
    }
}

// ---------------- GEMM2: out = h[1024,512] @ W2[512,1] + b2 (wave32 reduction/row)
__global__ __launch_bounds__(256)
void ncn_gemm2(const float* __restrict__ h, const float* __restrict__ W2,
               const float* __restrict__ b2, float* __restrict__ out, int B) {
    int wave = threadIdx.x >> 5;             // 8 waves per block, one row each
    int lane = threadIdx.x & 31;
    int row  = blockIdx.x * 8 + wave;
    if (row >= B) return;
    const float* hr = h + (size_t)row * DHID;
    float s = 0.0f;
    #pragma unroll
    for (int k = lane; k < DHID; k += 32) s += hr[k] * W2[k];
    #pragma unroll
    for (int off = 16; off > 0; off >>= 1) s += __shfl_xor(s, off, 32);
    if (lane == 0) out[row] = s + b2[0];
}

// ---------------------------------------------------------------------- launcher
extern "C" void kernel_launch(void* const* d_in, const int* in_sizes, int n_in,
                              void* d_out, int out_size, void* d_ws, size_t ws_size,
                              hipStream_t stream) {
    const float*     x   = (const float*)d_in[0];
    const long long* ei  = (const long long*)d_in[1];   // int64 [2, E]
    const long long* tar = (const long long*)d_in[2];   // int64 [2, B]
    const float*     W1  = (const float*)d_in[3];
    const float*     b1  = (const float*)d_in[4];
    const float*     W2  = (const float*)d_in[5];
    const float*     b2  = (const float*)d_in[6];

    const int N = in_sizes[0] / DIN;   // 100000
    const int E = in_sizes[1] / 2;     // 3200000
    const int B = in_sizes[2] / 2;     // 1024

    // workspace carve-out (~8 MB)
    int*   map_i  = (int*)d_ws;
    int*   map_j  = map_i + N;
    int*   cnt_i  = map_j + N;
    int*   cnt_j  = cnt_i + B;
    int*   list_i = cnt_j + B;
    int*   list_j = list_i + (size_t)B * CAP;
    float* xs     = (float*)(list_j + (size_t)B * CAP); // [B, 256] (8B-aligned)
    float* hbuf   = xs + (size_t)B * (2 * DIN);         // [B, 512]
    float* out    = (float*)d_out;

    const int T = 256;
    ncn_init   <<<(N + T - 1) / T, T, 0, stream>>>(map_i, map_j, cnt_i, cnt_j, N, B);
    ncn_setmap <<<(B + T - 1) / T, T, 0, stream>>>(tar, map_i, map_j, B);
    ncn_scatter<<<(E + T - 1) / T, T, 0, stream>>>(ei, E, map_i, map_j,
                                                   cnt_i, cnt_j, list_i, list_j);
    ncn_pairs  <<<B, 128, 0, stream>>>(x, tar, cnt_i, cnt_j, list_i, list_j, xs, B);

    dim3 g1(B / 16, DHID / 16);        // 64 x 32 tiles, one wave32 per tile
    ncn_gemm1_wmma<<<g1, 32, 0, stream>>>(xs, W1, b1, hbuf);
    ncn_gemm2<<<(B + 7) / 8, 256, 0, stream>>>(hbuf, W2, b2, out, B);
}